// YOLOX_29429115912382
// MI455X (gfx1250) — compile-verified
//
#include <hip/hip_runtime.h>
#include <math.h>

#define C_CLS   80
#define TOPK    1000
#define NB      16384        // buckets: float bits >> 17 (sign=0 -> < 16384)
#define BSHIFT  17
#define CAP     4096         // candidate capacity per level (32KB LDS sort)
#define NDET    3000
#define NPAD    3008
#define CONF_T  0.05f
#define NMS_THR 0.6f
#define OFF_V   100000.0f

typedef __attribute__((ext_vector_type(2))) float v2f;
typedef __attribute__((ext_vector_type(8))) float v8f;

__device__ __forceinline__ float sigmoidf_(float x) { return 1.0f / (1.0f + __expf(-x)); }

// ---------------- workspace layout (bytes) ----------------
#define OFF_HIST     0u                      // 3*NB u32 = 196608
#define OFF_CNT      196608u                 // 3 u32 (+pad)  (contiguous w/ hist for zeroing)
#define OFF_THR      196624u                 // 3 u32 (+pad)
#define OFF_CAND     196864u                 // 3*CAP u64 = 98304
#define OFF_DETBOX   295168u                 // NPAD*4 f32 = 48128
#define OFF_DETSC    343296u                 // NPAD f32
#define OFF_DETLB    355328u                 // NPAD i32
#define OFF_OFFBOX   367360u                 // NPAD*4 f32 (class-offset boxes)
#define OFF_AREA     415488u                 // NPAD f32
#define OFF_SSC      427520u                 // NPAD f32 (sorted scores)
#define OFF_IOU      439552u                 // NPAD*NPAD f32 = 36192256

__global__ void zero_u32(unsigned* p, int n) {
  int i = blockIdx.x * blockDim.x + threadIdx.x;
  if (i < n) p[i] = 0u;
}

// Pass 1: histogram of score-bits (positive float bits are order-preserving)
__global__ void hist_kernel(const float* __restrict__ obj, const float* __restrict__ cls,
                            int M, unsigned* __restrict__ hist) {
  int i = blockIdx.x * blockDim.x + threadIdx.x;
  if (i >= M) return;
  __builtin_prefetch(cls + i + 8192, 0, 1);   // global_prefetch_b8
  int a = i / C_CLS;
  float s = sqrtf(sigmoidf_(obj[a]) * sigmoidf_(cls[i]));
  atomicAdd(&hist[__float_as_uint(s) >> BSHIFT], 1u);
}

// Pass 2: find threshold bucket so that count(bucket >= T) >= TOPK
__global__ void thresh_kernel(const unsigned* __restrict__ hist, unsigned* __restrict__ thr) {
  if (threadIdx.x != 0) return;
  const unsigned* h = hist + blockIdx.x * NB;
  unsigned cum = 0; int b;
  for (b = NB - 1; b > 0; b--) { cum += h[b]; if (cum >= TOPK) break; }
  thr[blockIdx.x] = (unsigned)b;
}

// Pass 3: compact candidates (score_bits<<32 | ~idx : desc score, asc idx tiebreak)
__global__ void compact_kernel(const float* __restrict__ obj, const float* __restrict__ cls,
                               int M, const unsigned* __restrict__ thr,
                               unsigned* __restrict__ cnt, unsigned long long* __restrict__ cand) {
  int i = blockIdx.x * blockDim.x + threadIdx.x;
  if (i >= M) return;
  __builtin_prefetch(cls + i + 8192, 0, 1);
  int a = i / C_CLS;
  float s = sqrtf(sigmoidf_(obj[a]) * sigmoidf_(cls[i]));
  unsigned key = __float_as_uint(s);
  if ((key >> BSHIFT) >= *thr) {
    unsigned p = atomicAdd(cnt, 1u);
    if (p < CAP) cand[p] = ((unsigned long long)key << 32) | (unsigned)(~(unsigned)i);
  }
}

template <int S>
__device__ void bitonic_desc(unsigned long long* s) {
  int tid = threadIdx.x, nt = blockDim.x;
  for (int k = 2; k <= S; k <<= 1) {
    for (int j = k >> 1; j > 0; j >>= 1) {
      __syncthreads();
      for (int i = tid; i < S; i += nt) {
        int ixj = i ^ j;
        if (ixj > i) {
          unsigned long long x = s[i], y = s[ixj];
          bool dir = ((i & k) == 0);          // dir=true -> descending block
          if ((x < y) == dir) { s[i] = y; s[ixj] = x; }
        }
      }
    }
  }
  __syncthreads();
}

// Pass 4 (per level): sort candidates in LDS, take top-1000, decode boxes
__global__ void select_kernel(const unsigned long long* __restrict__ cand,
                              const unsigned* __restrict__ cnt,
                              const float* __restrict__ reg, int W, float stride, int base,
                              float* __restrict__ detBox, float* __restrict__ detScore,
                              int* __restrict__ detLabel) {
  __shared__ unsigned long long s[CAP];
  int tid = threadIdx.x;
  unsigned n = *cnt; if (n > CAP) n = CAP;
  for (int i = tid; i < CAP; i += blockDim.x) s[i] = (i < (int)n) ? cand[i] : 0ull;
  bitonic_desc<CAP>(s);
  if (tid < TOPK) {
    unsigned long long kv = s[tid];
    float score  = __uint_as_float((unsigned)(kv >> 32));
    unsigned idx = ~(unsigned)(kv & 0xFFFFFFFFu);
    int anchor = (int)(idx / C_CLS);
    int label  = (int)(idx % C_CLS);
    float ax = ((float)(anchor % W) + 0.5f) * stride;
    float ay = ((float)(anchor / W) + 0.5f) * stride;
    const float* r = reg + (size_t)anchor * 4;
    float cx = r[0] * stride + ax, cy = r[1] * stride + ay;
    float w  = __expf(r[2]) * stride, h = __expf(r[3]) * stride;
    int o = base + tid;
    detBox[o*4+0] = cx - 0.5f*w; detBox[o*4+1] = cy - 0.5f*h;
    detBox[o*4+2] = cx + 0.5f*w; detBox[o*4+3] = cy + 0.5f*h;
    detScore[o] = score; detLabel[o] = label;
  }
}

// Pass 5: global stable sort of the 3000 detections by score desc; emit boxes/labels,
// build class-offset boxes + areas for IoU.
__global__ void sort_all_kernel(const float* __restrict__ detBox, const float* __restrict__ detScore,
                                const int* __restrict__ detLabel,
                                float* __restrict__ outBoxes, float* __restrict__ outLabels,
                                float* __restrict__ offBox, float* __restrict__ area,
                                float* __restrict__ sScore) {
  __shared__ unsigned long long s[4096];
  int tid = threadIdx.x;
  for (int i = tid; i < 4096; i += blockDim.x) {
    if (i < NDET) s[i] = ((unsigned long long)__float_as_uint(detScore[i]) << 32) | (unsigned)(~(unsigned)i);
    else          s[i] = 0ull;
  }
  bitonic_desc<4096>(s);
  for (int t = tid; t < NPAD; t += blockDim.x) {
    if (t < NDET) {
      unsigned long long kv = s[t];
      unsigned pos = ~(unsigned)(kv & 0xFFFFFFFFu);
      float sc = __uint_as_float((unsigned)(kv >> 32));
      float x1 = detBox[pos*4+0], y1 = detBox[pos*4+1];
      float x2 = detBox[pos*4+2], y2 = detBox[pos*4+3];
      int lb = detLabel[pos];
      outBoxes[t*4+0] = x1; outBoxes[t*4+1] = y1; outBoxes[t*4+2] = x2; outBoxes[t*4+3] = y2;
      outLabels[t] = (float)lb;
      float off = (float)lb * OFF_V;
      float ox1 = x1 + off, oy1 = y1 + off, ox2 = x2 + off, oy2 = y2 + off;
      offBox[t*4+0] = ox1; offBox[t*4+1] = oy1; offBox[t*4+2] = ox2; offBox[t*4+3] = oy2;
      area[t] = fmaxf(ox2 - ox1, 0.0f) * fmaxf(oy2 - oy1, 0.0f);
      sScore[t] = sc;
    } else {
      offBox[t*4+0] = 0.f; offBox[t*4+1] = 0.f; offBox[t*4+2] = 0.f; offBox[t*4+3] = 0.f;
      area[t] = 0.f; sScore[t] = 0.f;
    }
  }
}

// Pass 6: 16x16 IoU tiles; pairwise (area_i + area_j) via V_WMMA_F32_16X16X4_F32 outer sum:
// A[m] = [area_i, 1, 0, 0], B[:,n] = [1, area_j, 0, 0]  =>  D[m][n] = area_i + area_j.
__global__ __launch_bounds__(32) void iou_kernel(const float* __restrict__ offBox,
                                                 const float* __restrict__ area,
                                                 float* __restrict__ iouMat) {
  int i0 = blockIdx.y * 16, j0 = blockIdx.x * 16;
  int lane = threadIdx.x;
  // A (16x4 f32): lanes 0-15 hold M=lane, VGPR0=K0, VGPR1=K1; lanes 16-31 hold K2/K3 (zero)
  v2f a, b;
  if (lane < 16) { a.x = area[i0 + lane]; a.y = 1.0f; b.x = 1.0f; b.y = area[j0 + lane]; }
  else           { a.x = 0.0f; a.y = 0.0f; b.x = 0.0f; b.y = 0.0f; }
  v8f c = {};
  v8f d = __builtin_amdgcn_wmma_f32_16x16x4_f32(false, a, false, b, (short)0, c, false, false);

  // D layout: lanes 0-15 -> N=lane, M=v; lanes 16-31 -> N=lane-16, M=8+v
  int n = lane & 15;
  int j = j0 + n;
  float jx1 = offBox[j*4+0], jy1 = offBox[j*4+1], jx2 = offBox[j*4+2], jy2 = offBox[j*4+3];
  int mbase = (lane < 16) ? 0 : 8;
#pragma unroll
  for (int v = 0; v < 8; v++) {
    int i = i0 + mbase + v;
    float ix1 = offBox[i*4+0], iy1 = offBox[i*4+1], ix2 = offBox[i*4+2], iy2 = offBox[i*4+3];
    float xx1 = fmaxf(ix1, jx1), yy1 = fmaxf(iy1, jy1);
    float xx2 = fminf(ix2, jx2), yy2 = fminf(iy2, jy2);
    float inter = fmaxf(xx2 - xx1, 0.0f) * fmaxf(yy2 - yy1, 0.0f);
    iouMat[i * NPAD + j] = inter / (d[v] - inter + 1e-10f);
  }
}

// Pass 7: serial NMS over sorted detections; emit final scores (0 where suppressed)
__global__ void nms_kernel(const float* __restrict__ iouMat, const float* __restrict__ sScore,
                           float* __restrict__ outScores) {
  __shared__ int keepL[NPAD];
  int tid = threadIdx.x;
  for (int i = tid; i < NPAD; i += blockDim.x)
    keepL[i] = (i < NDET) && (sScore[i] > CONF_T);
  __syncthreads();
  for (int i = 0; i < NDET; i++) {
    if (keepL[i]) {
      const float* row = iouMat + (size_t)i * NPAD;
      for (int j = i + 1 + tid; j < NDET; j += blockDim.x)
        if (row[j] > NMS_THR) keepL[j] = 0;
    }
    __syncthreads();
  }
  for (int t = tid; t < NDET; t += blockDim.x)
    outScores[t] = keepL[t] ? sScore[t] : 0.0f;
}

extern "C" void kernel_launch(void* const* d_in, const int* in_sizes, int n_in,
                              void* d_out, int out_size, void* d_ws, size_t ws_size,
                              hipStream_t stream) {
  const float* obj[3] = {(const float*)d_in[0], (const float*)d_in[3], (const float*)d_in[6]};
  const float* cls[3] = {(const float*)d_in[1], (const float*)d_in[4], (const float*)d_in[7]};
  const float* reg[3] = {(const float*)d_in[2], (const float*)d_in[5], (const float*)d_in[8]};
  const int   A[3]      = {65536, 16384, 4096};
  const int   W[3]      = {256, 128, 64};
  const float strd[3]   = {8.0f, 16.0f, 32.0f};

  char* ws = (char*)d_ws;
  unsigned*            hist     = (unsigned*)(ws + OFF_HIST);
  unsigned*            cnt      = (unsigned*)(ws + OFF_CNT);
  unsigned*            thr      = (unsigned*)(ws + OFF_THR);
  unsigned long long*  cand     = (unsigned long long*)(ws + OFF_CAND);
  float*               detBox   = (float*)(ws + OFF_DETBOX);
  float*               detScore = (float*)(ws + OFF_DETSC);
  int*                 detLabel = (int*)(ws + OFF_DETLB);
  float*               offBox   = (float*)(ws + OFF_OFFBOX);
  float*               area     = (float*)(ws + OFF_AREA);
  float*               sScore   = (float*)(ws + OFF_SSC);
  float*               iouMat   = (float*)(ws + OFF_IOU);

  // zero histograms + candidate counters every call (ws is poisoned, not re-zeroed)
  int nz = 3 * NB + 4;
  zero_u32<<<(nz + 255) / 256, 256, 0, stream>>>(hist, nz);

  for (int l = 0; l < 3; l++) {
    int M = A[l] * C_CLS;
    hist_kernel<<<(M + 255) / 256, 256, 0, stream>>>(obj[l], cls[l], M, hist + l * NB);
  }
  thresh_kernel<<<3, 32, 0, stream>>>(hist, thr);
  for (int l = 0; l < 3; l++) {
    int M = A[l] * C_CLS;
    compact_kernel<<<(M + 255) / 256, 256, 0, stream>>>(obj[l], cls[l], M, thr + l, cnt + l,
                                                        cand + (size_t)l * CAP);
  }
  for (int l = 0; l < 3; l++) {
    select_kernel<<<1, 1024, 0, stream>>>(cand + (size_t)l * CAP, cnt + l, reg[l], W[l], strd[l],
                                          l * TOPK, detBox, detScore, detLabel);
  }

  float* outB = (float*)d_out;          // [3000*4] boxes
  float* outS = outB + NDET * 4;        // [3000]   scores
  float* outL = outS + NDET;            // [3000]   labels (as float)

  sort_all_kernel<<<1, 1024, 0, stream>>>(detBox, detScore, detLabel, outB, outL,
                                          offBox, area, sScore);
  dim3 g(NPAD / 16, NPAD / 16);
  iou_kernel<<<g, 32, 0, stream>>>(offBox, area, iouMat);
  nms_kernel<<<1, 1024, 0, stream>>>(iouMat, sScore, outS);
}